// TFCriterion_43843026157892
// MI455X (gfx1250) — compile-verified
//
#include <hip/hip_runtime.h>

// Problem constants (match reference)
#define SEQ   256
#define BATCH 32
#define CDIM  64
#define NROWS (SEQ * BATCH)        // 8192 (l,b) pairs
#define ROWSTRIDE (BATCH * CDIM)   // 2048 floats per sequence step
#define TOT (NROWS * CDIM)         // 524288 floats per (L,B,C) tensor
#define MPAD 65                    // LDS row pitch (conflict-free both ways)

typedef __attribute__((ext_vector_type(2))) float v2f;
typedef __attribute__((ext_vector_type(4))) float v4f;
typedef __attribute__((ext_vector_type(8))) float v8f;

__device__ __forceinline__ v8f wmma_f32x4(v2f a, v2f b, v8f c) {
    return __builtin_amdgcn_wmma_f32_16x16x4_f32(false, a, false, b,
                                                 (short)0, c, false, false);
}

// ---------------------------------------------------------------------------
// Temporal mixing: out[s, bc] = sum_i W[s,i] * src[i, bc]
// which: 0=f_mp(past,f) 1=f_mf(fut,f) 2=s_mp(past,s) 3=s_mf(fut,s)
// One wave computes a 16x64 output strip = 4 N-tiles sharing one A-fragment
// (4 independent WMMA accumulator chains).  Decay weights are synthesized
// branch-free; EXEC stays all-1s as required by WMMA.
// ---------------------------------------------------------------------------
__global__ void __launch_bounds__(128)
temporal_wmma_kernel(const float* __restrict__ fIn,
                     const float* __restrict__ sIn,
                     float* __restrict__ ws) {
    int wid  = blockIdx.x * 4 + (threadIdx.x >> 5);  // 0..2047
    int lane = threadIdx.x & 31;
    int which = wid >> 9;            // 0..3  (512 waves per output tensor)
    int rem   = wid & 511;
    int st    = rem >> 5;            // 0..15 (s-tiles of 16)
    int ng    = rem & 31;            // 0..31 (N-groups of 64)
    int s0 = st * 16, n0 = ng * 64;

    const float* __restrict__ src = (which < 2) ? fIn : sIn;
    const bool usePast = ((which & 1) == 0);
    float* __restrict__ dst = ws + (size_t)which * TOT;

    int m  = lane & 15;
    int hi = lane >> 4;
    int srow = s0 + m;
    float inv = usePast ? (1.0f / fmaxf((float)srow, 1.0f))
                        : (1.0f / fmaxf((float)(SEQ - 1) - (float)srow, 1.0f));

    const float* srcp = src + n0 + m;
    v8f c0 = {}, c1 = {}, c2 = {}, c3 = {};

    for (int k0 = 0; k0 < SEQ; k0 += 4) {
        int ka = k0 + 2 * hi;
        float d0 = (float)(srow - ka);
        float d1 = (float)(srow - ka - 1);
        float w0 = __expf(-d0 * d0 * 0.125f) * inv;
        float w1 = __expf(-d1 * d1 * 0.125f) * inv;
        bool in0 = usePast ? (ka     < srow) : (ka     > srow);
        bool in1 = usePast ? (ka + 1 < srow) : (ka + 1 > srow);
        v2f a;
        a.x = in0 ? w0 : 0.0f;
        a.y = in1 ? w1 : 0.0f;

        const float* r0p = srcp + (size_t)ka * ROWSTRIDE;
        const float* r1p = r0p + ROWSTRIDE;
        v2f b;
        b.x = r0p[0];  b.y = r1p[0];  c0 = wmma_f32x4(a, b, c0);
        b.x = r0p[16]; b.y = r1p[16]; c1 = wmma_f32x4(a, b, c1);
        b.x = r0p[32]; b.y = r1p[32]; c2 = wmma_f32x4(a, b, c2);
        b.x = r0p[48]; b.y = r1p[48]; c3 = wmma_f32x4(a, b, c3);
    }
    #pragma unroll
    for (int vg = 0; vg < 8; ++vg) {
        int row = s0 + vg + 8 * hi;
        float* drow = dst + (size_t)row * ROWSTRIDE + n0 + m;
        drow[0]  = c0[vg];
        drow[16] = c1[vg];
        drow[32] = c2[vg];
        drow[48] = c3[vg];
    }
}

// ---------------------------------------------------------------------------
// Fused per-(l,b) kernel: 64 threads, one (l,b) each.
// Each 64x64 matrix streams global->LDS via CDNA5 async-LDS loads with a
// non-temporal hint (single-use 640MB stream must not thrash L2), into a
// pad-65 layout (staging writes and BOTH dot-product read patterns are
// bank-conflict-free, and all matrix ds_loads use literal offsets).
// LDS is DOUBLE-BUFFERED: tile k+1 streams (ASYNCcnt) while tile k computes;
// s_wait_asynccnt is deferred until after the dots.
// ---------------------------------------------------------------------------
__device__ __forceinline__ void stage_issue(const float* __restrict__ g,
                                            size_t base, float* mat, int tid) {
    const float* gp = g + base + tid;
    #pragma unroll
    for (int i = 0; i < 64; ++i) {
        unsigned lds_off = (unsigned)(size_t)&mat[i * MPAD + tid];
        const float* ga = gp + i * 64;
        asm volatile("global_load_async_to_lds_b32 %0, %1, off th:TH_LOAD_NT"
                     :: "v"(lds_off), "v"(ga) : "memory");
    }
}
__device__ __forceinline__ void wait_async() {
    asm volatile("s_wait_asynccnt 0x0" ::: "memory");
}

// cd[tid] = sum_r u[r]*M[r][tid]   (column dot)
// rd[tid] = sum_c M[tid][c]*w[c]   (row dot)
// u/w read as aligned float4 broadcasts; matrix reads are literal-offset b32.
__device__ __forceinline__ void dots(const float* mat, const float* u,
                                     const float* w, int tid,
                                     float& cd, float& rd) {
    float c0 = 0.0f, r0 = 0.0f;
    #pragma unroll
    for (int q = 0; q < 16; ++q) {
        v4f u4 = *(const v4f*)(u + 4 * q);
        v4f w4 = *(const v4f*)(w + 4 * q);
        #pragma unroll
        for (int j = 0; j < 4; ++j) {
            c0 += u4[j] * mat[(4 * q + j) * MPAD + tid];
            r0 += mat[tid * MPAD + 4 * q + j] * w4[j];
        }
    }
    cd = c0; rd = r0;
}

// Deterministic in-block softmax + CE.  Returns -log_softmax(x)[label]
// (same value in all threads).  Optionally writes softmax probs.
__device__ __forceinline__ float softmax_ce(float x, int label, int tid,
                                            float* red, float* xbuf,
                                            float* outp, size_t obase) {
    __syncthreads();                  // protect red/xbuf reuse across calls
    xbuf[tid] = x;
    red[tid]  = x;
    __syncthreads();
    #pragma unroll
    for (int sft = 32; sft > 0; sft >>= 1) {
        if (tid < sft) red[tid] = fmaxf(red[tid], red[tid + sft]);
        __syncthreads();
    }
    float m = red[0];
    __syncthreads();
    float e = expf(x - m);
    red[tid] = e;
    __syncthreads();
    #pragma unroll
    for (int sft = 32; sft > 0; sft >>= 1) {
        if (tid < sft) red[tid] += red[tid + sft];
        __syncthreads();
    }
    float Z = red[0];
    if (outp) outp[obase + tid] = e * __builtin_amdgcn_rcpf(Z);
    return m + logf(Z) - xbuf[label];
}

__global__ void __launch_bounds__(64)
fused_kernel(const float* __restrict__ fIn,  const float* __restrict__ sIn,
             const float* __restrict__ fsM,  const float* __restrict__ ffM,
             const float* __restrict__ ssM,  const float* __restrict__ fstM,
             const float* __restrict__ sftM,
             const int* __restrict__ f_labels, const int* __restrict__ s_labels,
             const float* __restrict__ ws_v,   // [f_mp | f_mf | s_mp | s_mf]
             float* __restrict__ part,
             float* __restrict__ out_f, float* __restrict__ out_s) {
    __shared__ __align__(16) float mats[2][64 * MPAD];
    __shared__ __align__(16) float vf[64], vs[64];
    __shared__ __align__(16) float vfmp[64], vfmf[64], vsmp[64], vsmf[64];
    __shared__ __align__(16) float red[64], xbuf[64];

    int lb  = blockIdx.x;            // l*B + b
    int tid = threadIdx.x;           // 0..63 = channel index
    size_t vo    = (size_t)lb * 64 + tid;
    size_t mbase = (size_t)lb * 4096;

    // Kick off first tile immediately; vector loads overlap the DMA.
    stage_issue(ffM, mbase, mats[0], tid);

    float fv = fIn[vo], sv = sIn[vo];
    vf[tid] = fv;                 vs[tid] = sv;
    vfmp[tid] = ws_v[vo];         vfmf[tid] = ws_v[(size_t)TOT + vo];
    vsmp[tid] = ws_v[2 * (size_t)TOT + vo];
    vsmf[tid] = ws_v[3 * (size_t)TOT + vo];

    wait_async();  __syncthreads();            // mats[0]=ff + vectors ready

    float acc_f = fv, acc_s = sv;
    float cd, rd;

    // --- ff (buf0): prefetch ss into buf1 while computing ---
    stage_issue(ssM, mbase, mats[1], tid);
    dots(mats[0], vfmp, vfmf, tid, cd, rd);
    acc_f += 0.5f * (cd + rd);
    wait_async();  __syncthreads();

    // --- ss (buf1): prefetch fs into buf0 ---
    stage_issue(fsM, mbase, mats[0], tid);
    dots(mats[1], vsmp, vsmf, tid, cd, rd);
    acc_s += 0.5f * (cd + rd);
    wait_async();  __syncthreads();

    // --- fs (buf0): prefetch fs_t into buf1 ---
    stage_issue(fstM, mbase, mats[1], tid);
    dots(mats[0], vf, vs, tid, cd, rd);
    acc_f += 0.5f * rd;  acc_s += 0.5f * cd;
    wait_async();  __syncthreads();

    // --- fs_t (buf1): prefetch sf_t into buf0 ---
    stage_issue(sftM, mbase, mats[0], tid);
    dots(mats[1], vfmp, vsmf, tid, cd, rd);
    acc_f += 0.5f * rd;  acc_s += 0.5f * cd;
    wait_async();  __syncthreads();

    // --- sf_t (buf0) ---
    dots(mats[0], vsmp, vfmf, tid, cd, rd);
    acc_f += 0.5f * cd;  acc_s += 0.5f * rd;

    // loss = CE(f) + CE(s) + CE(next_f) + CE(next_s); outputs = softmax(next)
    int flab = f_labels[lb], slab = s_labels[lb];
    float nls = 0.0f;
    nls += softmax_ce(fv,    flab, tid, red, xbuf, nullptr, 0);
    nls += softmax_ce(sv,    slab, tid, red, xbuf, nullptr, 0);
    nls += softmax_ce(acc_f, flab, tid, red, xbuf, out_f, (size_t)lb * 64);
    nls += softmax_ce(acc_s, slab, tid, red, xbuf, out_s, (size_t)lb * 64);
    if (tid == 0) part[lb] = nls;
}

// ---------------------------------------------------------------------------
// Deterministic final reduction of 8192 per-block loss partials.
// ---------------------------------------------------------------------------
__global__ void __launch_bounds__(256)
finalize_kernel(const float* __restrict__ part, float* __restrict__ out_loss) {
    __shared__ float red[256];
    int tid = threadIdx.x;
    float acc = 0.0f;
    #pragma unroll
    for (int i = 0; i < 32; ++i) acc += part[tid + 256 * i];
    red[tid] = acc;
    __syncthreads();
    #pragma unroll
    for (int sft = 128; sft > 0; sft >>= 1) {
        if (tid < sft) red[tid] += red[tid + sft];
        __syncthreads();
    }
    if (tid == 0) out_loss[0] = red[0] / 8192.0f;
}

extern "C" void kernel_launch(void* const* d_in, const int* in_sizes, int n_in,
                              void* d_out, int out_size, void* d_ws, size_t ws_size,
                              hipStream_t stream) {
    const float* f     = (const float*)d_in[0];
    const float* s     = (const float*)d_in[1];
    const float* fs    = (const float*)d_in[2];
    const float* ff    = (const float*)d_in[3];
    const float* ss    = (const float*)d_in[4];
    const float* fs_t  = (const float*)d_in[5];
    const float* sf_t  = (const float*)d_in[6];
    const int* f_labels = (const int*)d_in[7];
    const int* s_labels = (const int*)d_in[8];

    float* out = (float*)d_out;          // [f_out | s_out | loss]
    float* ws  = (float*)d_ws;           // [f_mp|f_mf|s_mp|s_mf|partials]
    float* part = ws + 4 * (size_t)TOT;

    temporal_wmma_kernel<<<512, 128, 0, stream>>>(f, s, ws);
    fused_kernel<<<8192, 64, 0, stream>>>(f, s, fs, ff, ss, fs_t, sf_t,
                                          f_labels, s_labels, ws,
                                          part, out, out + TOT);
    finalize_kernel<<<1, 256, 0, stream>>>(part, out + 2 * (size_t)TOT);
}